// local_emb_D_1005022347289
// MI455X (gfx1250) — compile-verified
//
#include <hip/hip_runtime.h>

typedef __attribute__((ext_vector_type(2))) float v2f;
typedef __attribute__((ext_vector_type(4))) float v4f;
typedef __attribute__((ext_vector_type(8))) float v8f;

#define H       128
#define ROWPAD  132          // 132 mod 64 == 4 -> conflict-free bank rotation
#define TILE    16
#define K2_WAVES 2           // waves (tiles) per block in the edge kernel

// ---------------------------------------------------------------------------
// Kernel 1: L2-normalize rows. One wave32 per row (128 floats = float4/lane).
// emb is read once -> nontemporal load; e is written with RT policy so the
// 51.2MB table parks in the 192MB L2 for the gather kernel.
// ---------------------------------------------------------------------------
__global__ void normalize_rows_k(const float* __restrict__ emb,
                                 float* __restrict__ e, int n_nodes) {
    int wave = threadIdx.x >> 5;
    int lane = threadIdx.x & 31;
    int row  = blockIdx.x * (blockDim.x >> 5) + wave;
    if (row >= n_nodes) return;

    const v4f* rp = (const v4f*)(emb + (size_t)row * H);
    v4f v = __builtin_nontemporal_load(&rp[lane]);

    float s = v.x * v.x + v.y * v.y + v.z * v.z + v.w * v.w;
#pragma unroll
    for (int off = 16; off > 0; off >>= 1)
        s += __shfl_xor(s, off, 32);

    float inv = 1.0f / fmaxf(sqrtf(s), 1e-12f);   // x / max(||x||, eps)
    v4f o;
    o.x = v.x * inv; o.y = v.y * inv; o.z = v.z * inv; o.w = v.w * inv;
    ((v4f*)(e + (size_t)row * H))[lane] = o;
}

// ---------------------------------------------------------------------------
// Kernel 2: 16 edges per wave. Stage 16 src rows + 16 dst rows into LDS with
// fully-coalesced wave-wide float4 loads (one 512B row per iteration), then
// run 32 chained V_WMMA_F32_16X16X4_F32 ops; z[i] = D[i][i] * scale.
// A fragment is pre-scaled by d (ew = e * d). Full f32 precision.
// ---------------------------------------------------------------------------
__global__ void edge_dot_k(const float* __restrict__ e,
                           const float* __restrict__ dvec,
                           const float* __restrict__ scale,
                           const int*   __restrict__ src,
                           const int*   __restrict__ dst,
                           float*       __restrict__ z,
                           int n_edges) {
    __shared__ float lds[K2_WAVES * 32 * ROWPAD];

    int wave = threadIdx.x >> 5;
    int lane = threadIdx.x & 31;
    long long tile = (long long)blockIdx.x * K2_WAVES + wave;
    long long base = tile * TILE;
    if (base >= n_edges) return;                 // wave-uniform: EXEC stays all-1s

    int m = lane & 15;                           // edge-in-tile this lane serves
    long long eidx = base + m;
    if (eidx >= n_edges) eidx = n_edges - 1;     // tail clamp (stores guarded below)

    // lanes 0..15 carry src indices, lanes 16..31 carry dst indices
    int rowidx = (lane < 16) ? __builtin_nontemporal_load(&src[eidx])
                             : __builtin_nontemporal_load(&dst[eidx]);

    float* buf = lds + wave * (32 * ROWPAD);
#pragma unroll 8
    for (int r = 0; r < 32; ++r) {               // rows 0..15: src, 16..31: dst
        int ri = __shfl(rowidx, r, 32);
        v4f v = ((const v4f*)(e + (size_t)ri * H))[lane];
        *(v4f*)(buf + r * ROWPAD + 4 * lane) = v;
    }

    // WMMA fragment addressing (f32 16x16x4): lane L -> row L%16, K pair at
    // 2*(L/16). Two accumulators to decouple consecutive WMMA chains.
    int koff = (lane >> 4) << 1;                 // 0 or 2
    const float* arow = buf + m * ROWPAD;        // src row (A operand)
    const float* brow = buf + (16 + m) * ROWPAD; // dst row (B operand)

    v8f acc0 = {};
    v8f acc1 = {};
#pragma unroll
    for (int k0 = 0; k0 < H; k0 += 8) {
        v2f a0 = *(const v2f*)(arow + k0 + koff);
        v2f b0 = *(const v2f*)(brow + k0 + koff);
        v2f d0 = *(const v2f*)(dvec + k0 + koff);
        a0 *= d0;                                // fold in diag(d) -> ew
        acc0 = __builtin_amdgcn_wmma_f32_16x16x4_f32(
            false, a0, false, b0, (short)0, acc0, false, false);

        v2f a1 = *(const v2f*)(arow + k0 + 4 + koff);
        v2f b1 = *(const v2f*)(brow + k0 + 4 + koff);
        v2f d1 = *(const v2f*)(dvec + k0 + 4 + koff);
        a1 *= d1;
        acc1 = __builtin_amdgcn_wmma_f32_16x16x4_f32(
            false, a1, false, b1, (short)0, acc1, false, false);
    }
    v8f acc = acc0 + acc1;

    // Diagonal lives in lanes 0..7 (VGPR=lane) and lanes 24..31 (VGPR=lane-24).
    int r = lane & 7;
    float dv = acc[0];
    dv = (r == 1) ? acc[1] : dv;
    dv = (r == 2) ? acc[2] : dv;
    dv = (r == 3) ? acc[3] : dv;
    dv = (r == 4) ? acc[4] : dv;
    dv = (r == 5) ? acc[5] : dv;
    dv = (r == 6) ? acc[6] : dv;
    dv = (r == 7) ? acc[7] : dv;

    bool active = (lane < 8) || (lane >= 24);
    int  mm     = (lane < 8) ? lane : (lane - 16);
    long long oe = base + mm;
    float sc = scale[0];
    if (active && oe < n_edges)
        __builtin_nontemporal_store(dv * sc, &z[oe]);
}

// ---------------------------------------------------------------------------
extern "C" void kernel_launch(void* const* d_in, const int* in_sizes, int n_in,
                              void* d_out, int out_size, void* d_ws, size_t ws_size,
                              hipStream_t stream) {
    const float* emb   = (const float*)d_in[0];
    const float* dvec  = (const float*)d_in[1];
    const float* scale = (const float*)d_in[2];
    const int*   src   = (const int*)d_in[3];
    const int*   dst   = (const int*)d_in[4];
    float*       z     = (float*)d_out;

    int n_nodes = in_sizes[0] / H;
    int n_edges = in_sizes[3];

    float* e = (float*)d_ws;                     // normalized table, 51.2 MB

    // Kernel 1: 8 rows per 256-thread block (8 waves)
    int rows_per_blk = 256 / 32;
    int nb1 = (n_nodes + rows_per_blk - 1) / rows_per_blk;
    normalize_rows_k<<<nb1, 256, 0, stream>>>(emb, e, n_nodes);

    // Kernel 2: 16 edges per wave, K2_WAVES waves per block
    long long tiles = ((long long)n_edges + TILE - 1) / TILE;
    int nb2 = (int)((tiles + K2_WAVES - 1) / K2_WAVES);
    edge_dot_k<<<nb2, K2_WAVES * 32, 0, stream>>>(e, dvec, scale, src, dst, z,
                                                  n_edges);
}